// ST_RGCN_22067541967679
// MI455X (gfx1250) — compile-verified
//
#include <hip/hip_runtime.h>
#include <hip/hip_bf16.h>

// CDNA5 / gfx1250: wave32 only, WMMA fp32 path V_WMMA_F32_16X16X4_F32.
typedef __attribute__((ext_vector_type(2))) float v2f;
typedef __attribute__((ext_vector_type(8))) float v8f;

#define LDS_STRIDE 260   // 64-row tiles, stride 260 dwords -> bank-conflict-free column reads
#define NODES 64
#define CH 256
#define KPAIRS_MAT (CH / 2 * CH)     // 32768 float2 per packed 256x256 matrix

__device__ __forceinline__ v8f wmma_f32(v2f a, v2f b, v8f c) {
    // 8-arg form: (neg_a, A, neg_b, B, c_mod, C, reuse_a, reuse_b)
    return __builtin_amdgcn_wmma_f32_16x16x4_f32(false, a, false, b, (short)0, c,
                                                 false, false);
}

// A fragment (16x4 fp32): lane L<16 holds row base+L at K = k..k+1 in v.x/v.y,
// lanes 16..31 hold K = k+2..k+3 (ISA 32-bit A layout).
__device__ __forceinline__ v2f frag_a(const float* __restrict__ src, int stride,
                                      int rowBase, int k, int lane) {
    const int row = rowBase + (lane & 15);
    const int kk  = k + ((lane >> 4) << 1);
    v2f a;
    a.x = src[row * stride + kk];
    a.y = src[row * stride + kk + 1];
    return a;
}

// B fragment (4x16 fp32) from LDS (row-major, two strided dword reads).
__device__ __forceinline__ v2f frag_b_lds(const float* __restrict__ src, int stride,
                                          int k, int colBase, int lane) {
    const int col = colBase + (lane & 15);
    const int kk  = k + ((lane >> 4) << 1);
    v2f b;
    b.x = src[kk * stride + col];
    b.y = src[(kk + 1) * stride + col];
    return b;
}

// B fragment from pre-packed weights: Wp[kp*256 + col] = {W[2kp][col], W[2kp+1][col]}
// -> one global_load_b64 per lane, lanes 0-15 contiguous 128B.
__device__ __forceinline__ v2f frag_bp(const float2* __restrict__ Wp, int k,
                                       int colBase, int lane) {
    const int idx = ((k >> 1) + (lane >> 4)) * CH + colBase + (lane & 15);
    const float2 v = Wp[idx];
    v2f b;
    b.x = v.x;
    b.y = v.y;
    return b;
}

__device__ __forceinline__ v8f zero8() {
    v8f z = {0.f, 0.f, 0.f, 0.f, 0.f, 0.f, 0.f, 0.f};
    return z;
}

// ---------------------------------------------------------------------------
// Kernel 0: pack all weight matrices into WMMA-B-fragment-native float2 pairs.
//   wselfP : 1 matrix   (w_self[ci][co], pairs along ci)
//   wrelP  : 4 matrices (w_rel[r][ci][co])
//   wtP    : 9 matrices (conv tap k, transposed to [ci][co] on the fly)
// ---------------------------------------------------------------------------
__global__ void pack_weights_kernel(const float* __restrict__ w_self,
                                    const float* __restrict__ w_rel,
                                    const float* __restrict__ w_t,
                                    float2* __restrict__ wselfP,
                                    float2* __restrict__ wrelP,
                                    float2* __restrict__ wtP) {
    int i = blockIdx.x * blockDim.x + threadIdx.x;
    if (i < KPAIRS_MAT) {
        int kp = i >> 8, col = i & 255;
        wselfP[i] = make_float2(w_self[(2 * kp) * CH + col],
                                w_self[(2 * kp + 1) * CH + col]);
    } else if (i < 5 * KPAIRS_MAT) {
        int j = i - KPAIRS_MAT;
        int r = j >> 15, m = j & (KPAIRS_MAT - 1);
        int kp = m >> 8, col = m & 255;
        const float* W = w_rel + r * CH * CH;
        wrelP[j] = make_float2(W[(2 * kp) * CH + col], W[(2 * kp + 1) * CH + col]);
    } else if (i < 14 * KPAIRS_MAT) {
        int j = i - 5 * KPAIRS_MAT;
        int k = j >> 15, m = j & (KPAIRS_MAT - 1);
        int kp = m >> 8, col = m & 255;
        // W_k[ci][co] = w_t[co][ci][k]  (w_t is (Cout, Cin, K, 1))
        wtP[j] = make_float2(w_t[col * (CH * 9) + (2 * kp) * 9 + k],
                             w_t[col * (CH * 9) + (2 * kp + 1) * 9 + k]);
    }
}

// ---------------------------------------------------------------------------
// Kernel 1: n = relu(x@W_self + sum_r (A_r@x)@W_r + x), one block per time step.
// Dynamic LDS: Msm[64*260] | Psm[64*260]  (130 KB of the 320 KB WGP pool).
// ---------------------------------------------------------------------------
__global__ __launch_bounds__(256) void st_rgcn_msg_kernel(
    const float* __restrict__ x, const float* __restrict__ rel,
    const float2* __restrict__ wselfP, const float2* __restrict__ wrelP,
    float* __restrict__ nbuf) {
    extern __shared__ float smem[];
    float* Msm = smem;                       // x[t], 64 x 256 (stride 260)
    float* Psm = smem + NODES * LDS_STRIDE;  // A_r @ M staging

    const int t    = blockIdx.x;
    const int tid  = threadIdx.x;
    const int lane = tid & 31;
    const int wave = tid >> 5;   // 8 wave32 per block
    const int mt   = wave & 3;   // which 16-row strip of the 64 output rows
    const int g    = wave >> 2;  // which 128-col half

    // Stage x[t] into LDS (float4, padded stride).
    {
        const float4* src = (const float4*)(x + (size_t)t * NODES * CH);
        for (int i = tid; i < NODES * (CH / 4); i += 256) {
            float4 v   = src[i];
            float* dst = &Msm[(i >> 6) * LDS_STRIDE + ((i & 63) << 2)];
            dst[0] = v.x; dst[1] = v.y; dst[2] = v.z; dst[3] = v.w;
        }
    }
    __syncthreads();

    v8f acc[8];
#pragma unroll
    for (int i = 0; i < 8; ++i) acc[i] = zero8();

    // Self term: acc += M @ W_self  (packed B streams as b64 from L2)
    for (int k = 0; k < CH; k += 4) {
        v2f a = frag_a(Msm, LDS_STRIDE, mt * 16, k, lane);
#pragma unroll
        for (int i = 0; i < 8; ++i) {
            v2f bb = frag_bp(wselfP, k, (g * 8 + i) * 16, lane);
            acc[i] = wmma_f32(a, bb, acc[i]);
        }
    }

    // Relation terms: acc += (A_r @ M) @ W_r
    for (int r = 0; r < 4; ++r) {
        v8f p[8];
#pragma unroll
        for (int i = 0; i < 8; ++i) p[i] = zero8();

        const float* Ar = rel + r * NODES * NODES;  // rel[0][r]
        for (int k = 0; k < NODES; k += 4) {
            v2f a = frag_a(Ar, NODES, mt * 16, k, lane);
#pragma unroll
            for (int i = 0; i < 8; ++i) {
                v2f bb = frag_b_lds(Msm, LDS_STRIDE, k, (g * 8 + i) * 16, lane);
                p[i] = wmma_f32(a, bb, p[i]);
            }
        }

        __syncthreads();  // Psm readers of previous r are done
#pragma unroll
        for (int i = 0; i < 8; ++i) {
            const int col = (g * 8 + i) * 16 + (lane & 15);
            const int rb  = mt * 16 + ((lane >> 4) << 3);  // D layout: +8 rows for hi lanes
#pragma unroll
            for (int j = 0; j < 8; ++j) Psm[(rb + j) * LDS_STRIDE + col] = p[i][j];
        }
        __syncthreads();

        const float2* WrP = wrelP + r * KPAIRS_MAT;
        for (int k = 0; k < CH; k += 4) {
            v2f a = frag_a(Psm, LDS_STRIDE, mt * 16, k, lane);
#pragma unroll
            for (int i = 0; i < 8; ++i) {
                v2f bb = frag_bp(WrP, k, (g * 8 + i) * 16, lane);
                acc[i] = wmma_f32(a, bb, acc[i]);
            }
        }
    }

    // Epilogue: n = relu(acc + x)
    float* nb = nbuf + (size_t)t * NODES * CH;
#pragma unroll
    for (int i = 0; i < 8; ++i) {
        const int col = (g * 8 + i) * 16 + (lane & 15);
        const int rb  = mt * 16 + ((lane >> 4) << 3);
#pragma unroll
        for (int j = 0; j < 8; ++j) {
            float v = acc[i][j] + Msm[(rb + j) * LDS_STRIDE + col];
            nb[(rb + j) * CH + col] = v > 0.0f ? v : 0.0f;
        }
    }
}

// ---------------------------------------------------------------------------
// Kernel 2: temporal conv (K=9, pad 4, per-segment zero pad) + bias.
// z[t] = sum_k n[t+k-4] @ W_k + b   as 9 accumulated GEMMs.
// Also emits per-block per-channel (sum, sumsq) partials (deterministic).
// Dynamic LDS: Nsm[64*260]; static: red[512].
// ---------------------------------------------------------------------------
__global__ __launch_bounds__(256) void tconv_kernel(
    const float* __restrict__ nbuf, const float2* __restrict__ wtP,
    const float* __restrict__ b_t, const int* __restrict__ xlen,
    float* __restrict__ zout, float* __restrict__ part) {
    extern __shared__ float smem[];
    float* Nsm = smem;
    __shared__ float red[2 * CH];

    const int t    = blockIdx.x;
    const int tid  = threadIdx.x;
    const int lane = tid & 31;
    const int wave = tid >> 5;
    const int mt   = wave & 3;
    const int g    = wave >> 2;

    // Segment bounds [s0, s1) containing t (block-uniform).
    int s0 = 0, s1 = 0, off = 0;
    for (int s = 0; s < 8; ++s) {
        int l = xlen[s];
        if (t >= off && t < off + l) { s0 = off; s1 = off + l; }
        off += l;
    }

    red[tid] = 0.0f;
    red[tid + CH] = 0.0f;
    __syncthreads();

    v8f acc[8];
#pragma unroll
    for (int i = 0; i < 8; ++i) acc[i] = zero8();

    for (int k = 0; k < 9; ++k) {
        const int ts = t + k - 4;
        if (ts >= s0 && ts < s1) {  // block-uniform: barriers inside are legal
            __syncthreads();
            const float4* src = (const float4*)(nbuf + (size_t)ts * NODES * CH);
            for (int i = tid; i < NODES * (CH / 4); i += 256) {
                float4 v   = src[i];
                float* dst = &Nsm[(i >> 6) * LDS_STRIDE + ((i & 63) << 2)];
                dst[0] = v.x; dst[1] = v.y; dst[2] = v.z; dst[3] = v.w;
            }
            __syncthreads();

            const float2* WkP = wtP + k * KPAIRS_MAT;
            for (int kk = 0; kk < CH; kk += 4) {
                v2f a = frag_a(Nsm, LDS_STRIDE, mt * 16, kk, lane);
#pragma unroll
                for (int i = 0; i < 8; ++i) {
                    v2f bb = frag_bp(WkP, kk, (g * 8 + i) * 16, lane);
                    acc[i] = wmma_f32(a, bb, acc[i]);
                }
            }
        }
    }

    // Epilogue: bias, store pre-norm z, per-channel LDS reduction (ds_add_f32).
    float* zb = zout + (size_t)t * NODES * CH;
#pragma unroll
    for (int i = 0; i < 8; ++i) {
        const int col  = (g * 8 + i) * 16 + (lane & 15);
        const int rb   = mt * 16 + ((lane >> 4) << 3);
        const float bc = b_t[col];
        float s = 0.0f, s2 = 0.0f;
#pragma unroll
        for (int j = 0; j < 8; ++j) {
            float v = acc[i][j] + bc;
            zb[(rb + j) * CH + col] = v;
            s += v;
            s2 += v * v;
        }
        atomicAdd(&red[col], s);
        atomicAdd(&red[CH + col], s2);
    }
    __syncthreads();
    part[(size_t)t * 512 + tid]       = red[tid];
    part[(size_t)t * 512 + 256 + tid] = red[tid + CH];
}

// ---------------------------------------------------------------------------
// Kernel 3: deterministic reduction of partials -> scale/shift per channel.
// ---------------------------------------------------------------------------
__global__ void stats_kernel(const float* __restrict__ part,
                             const float* __restrict__ gamma,
                             const float* __restrict__ beta,
                             float* __restrict__ scsh, int T) {
    const int c = threadIdx.x;  // 256 threads, 1 block
    float s = 0.0f, s2 = 0.0f;
    for (int b = 0; b < T; ++b) {
        s  += part[(size_t)b * 512 + c];
        s2 += part[(size_t)b * 512 + 256 + c];
    }
    const float invN = 1.0f / (float)(T * NODES);
    const float mean = s * invN;
    const float var  = s2 * invN - mean * mean;
    const float rstd = rsqrtf(var + 1e-5f);
    const float sc   = gamma[c] * rstd;
    scsh[c]      = sc;
    scsh[CH + c] = beta[c] - mean * sc;
}

// ---------------------------------------------------------------------------
// Kernel 4: out = relu(z*scale + shift + x), float4; tail = new_lens (= lens).
// ---------------------------------------------------------------------------
__global__ void out_kernel(const float* __restrict__ x,
                           const float* __restrict__ scsh,
                           const int* __restrict__ xlen,
                           float* __restrict__ out, int total4) {
    int i = blockIdx.x * blockDim.x + threadIdx.x;
    if (blockIdx.x == 0 && threadIdx.x < 8) {
        // new_lens[l] = (l + 2*PAD - K + STRIDE)/STRIDE = l  -> copy xlen bits
        ((int*)out)[(size_t)total4 * 4 + threadIdx.x] = xlen[threadIdx.x];
    }
    if (i < total4) {
        float4 z  = ((const float4*)out)[i];
        float4 xi = ((const float4*)x)[i];
        int c = (i << 2) & 255;  // channel of component .x (CH divisible by 4)
        float4 r;
        r.x = z.x * scsh[c + 0] + scsh[CH + c + 0] + xi.x;
        r.y = z.y * scsh[c + 1] + scsh[CH + c + 1] + xi.y;
        r.z = z.z * scsh[c + 2] + scsh[CH + c + 2] + xi.z;
        r.w = z.w * scsh[c + 3] + scsh[CH + c + 3] + xi.w;
        r.x = r.x > 0.f ? r.x : 0.f;
        r.y = r.y > 0.f ? r.y : 0.f;
        r.z = r.z > 0.f ? r.z : 0.f;
        r.w = r.w > 0.f ? r.w : 0.f;
        ((float4*)out)[i] = r;
    }
}

// ---------------------------------------------------------------------------
extern "C" void kernel_launch(void* const* d_in, const int* in_sizes, int n_in,
                              void* d_out, int out_size, void* d_ws,
                              size_t ws_size, hipStream_t stream) {
    const float* x      = (const float*)d_in[0];
    const float* rel    = (const float*)d_in[1];
    const int*   xlen   = (const int*)d_in[2];
    const float* w_self = (const float*)d_in[3];
    const float* w_rel  = (const float*)d_in[4];
    const float* w_t    = (const float*)d_in[5];
    const float* b_t    = (const float*)d_in[6];
    const float* gamma  = (const float*)d_in[7];
    const float* beta   = (const float*)d_in[8];

    const int T     = in_sizes[0] / (NODES * CH);  // 1536
    const int total = T * NODES * CH;
    float* out = (float*)d_out;

    // Workspace layout (floats): nbuf | wselfP | wrelP | wtP | part | scsh
    float*  nbuf   = (float*)d_ws;
    float2* wselfP = (float2*)(nbuf + (size_t)total);
    float2* wrelP  = wselfP + KPAIRS_MAT;
    float2* wtP    = wrelP + 4 * KPAIRS_MAT;
    float*  part   = (float*)(wtP + 9 * KPAIRS_MAT);
    float*  scsh   = part + (size_t)T * 512;

    const size_t smem1 = (size_t)(2 * NODES * LDS_STRIDE) * sizeof(float);  // 130 KB
    const size_t smem2 = (size_t)(NODES * LDS_STRIDE) * sizeof(float);      // 65 KB

    pack_weights_kernel<<<(14 * KPAIRS_MAT + 255) / 256, 256, 0, stream>>>(
        w_self, w_rel, w_t, wselfP, wrelP, wtP);
    st_rgcn_msg_kernel<<<T, 256, smem1, stream>>>(x, rel, wselfP, wrelP, nbuf);
    tconv_kernel<<<T, 256, smem2, stream>>>(nbuf, wtP, b_t, xlen, out, part);
    stats_kernel<<<1, 256, 0, stream>>>(part, gamma, beta, scsh, T);
    out_kernel<<<(total / 4 + 255) / 256, 256, 0, stream>>>(x, scsh, xlen, out,
                                                            total / 4);
}